// AccFlow_16836271800625
// MI455X (gfx1250) — compile-verified
//
#include <hip/hip_runtime.h>

typedef float v2f __attribute__((ext_vector_type(2)));
typedef float v8f __attribute__((ext_vector_type(8)));
typedef unsigned long long u64;

#define KNN 8

// Sentinel key: d2 = +inf (so any finite candidate beats it via float compare),
// index = all-ones.
#define KEY_SENTINEL 0x7F800000FFFFFFFFull

// ---------------------------------------------------------------------------
// Pack refs into WMMA A-fragment-friendly form: pk[m] = (-2rx, -2ry, -2rz, |r|^2)
// ---------------------------------------------------------------------------
__global__ void pack_refs_kernel(const float* __restrict__ ref,
                                 float4* __restrict__ pk, int M) {
  int m = blockIdx.x * blockDim.x + threadIdx.x;
  if (m < M) {
    float x = ref[3 * m + 0], y = ref[3 * m + 1], z = ref[3 * m + 2];
    pk[m] = make_float4(-2.0f * x, -2.0f * y, -2.0f * z, x * x + y * y + z * z);
  }
}

// Unguarded insert: caller has already established key < kl[KNN-1].
// Single u64 value array -> lowers to a clean v_min_u64/v_max_u64 chain.
__device__ __forceinline__ void insert8_hit(u64 (&kl)[KNN], u64 key) {
  kl[KNN - 1] = key;
#pragma unroll
  for (int j = KNN - 1; j >= 1; --j) {
    u64 a = kl[j - 1];
    u64 b = kl[j];
    u64 lo = (b < a) ? b : a;
    u64 hi = (b < a) ? a : b;
    kl[j - 1] = lo;
    kl[j]     = hi;
  }
}

// Exact guarded insert (used in the small merge phase only).
__device__ __forceinline__ void insert8(u64 (&kl)[KNN], u64 key) {
  if (key < kl[KNN - 1]) insert8_hit(kl, key);
}

// ---------------------------------------------------------------------------
// One wave handles 16 queries. Packed refs (M*16B = 256KB) staged once per
// block into LDS (fits CDNA5's 320KB WGP LDS). Ref loop: one
// V_WMMA_F32_16X16X4_F32 per 16-ref tile, A-fragment prefetched one tile
// ahead via conflict-free ds_load_b64 so DS latency hides under WMMA+select.
// Selection: per-lane register top-8 of u64 keys (d2 bits << 32 | index),
// float-threshold prefilter in the hot loop, exact u64 order inside.
// ---------------------------------------------------------------------------
__global__ void __launch_bounds__(256)
idw_knn_wmma_kernel(const float* __restrict__ qpts,
                    const float4* __restrict__ pk,
                    const float* __restrict__ rflow,
                    float* __restrict__ out, int N, int M) {
  extern __shared__ float4 spk[];  // M float4s + 1 pad tile

  // ---- stage packed refs into LDS (coalesced b128 -> ds_store_b128) ----
  for (int i = threadIdx.x; i < M; i += blockDim.x) spk[i] = pk[i];
  __syncthreads();

  const int lane   = threadIdx.x & 31;
  const int waveId = (int)((blockIdx.x * blockDim.x + threadIdx.x) >> 5);
  const int col    = lane & 15;
  const int hi     = lane >> 4;          // 0: supplies K=0,1   1: supplies K=2,3
  const int qbase  = waveId * 16;
  if (qbase >= N) return;                // whole-wave uniform (keeps EXEC full)
  const int q = qbase + col;

  // Query point / |q|^2 / B fragment: col n = [qx, qy, qz, 1]
  const float qx = qpts[3 * q + 0];
  const float qy = qpts[3 * q + 1];
  const float qz = qpts[3 * q + 2];
  const float q2 = qx * qx + qy * qy + qz * qz;

  v2f b;
  b.x = hi ? qz : qx;
  b.y = hi ? 1.0f : qy;

  u64 kl[KNN];
#pragma unroll
  for (int j = 0; j < KNN; ++j) kl[j] = KEY_SENTINEL;

  const int mofs  = hi * 8;  // C rows this lane observes: v + mofs
  const int tiles = M >> 4;

  // Lane's A-fragment stream: hi==0 -> (-2rx,-2ry) ; hi==1 -> (-2rz,|r|^2).
  // Addresses stride 16B across cols (+8B for hi half): all 64 banks, no
  // conflicts. One 256B tile per wave per iteration, prefetched a tile ahead.
  const float2* __restrict__ p = (const float2*)spk + (2 * col + hi);
  float2 av = *p;

#pragma unroll 2
  for (int t = 0; t < tiles; ++t) {
    float2 nx = p[32];  // next tile (pad tile on last iteration)
    p += 32;

    v2f a;
    a.x = av.x;
    a.y = av.y;

    v8f c = {0.f, 0.f, 0.f, 0.f, 0.f, 0.f, 0.f, 0.f};
    c = __builtin_amdgcn_wmma_f32_16x16x4_f32(
        /*neg_a=*/false, a, /*neg_b=*/false, b,
        /*c_mod=*/(short)0, c, /*reuse_a=*/false, /*reuse_b=*/false);

    const int mbase = t * 16 + mofs;
#pragma unroll
    for (int v = 0; v < 8; ++v) {
      float d2c = fmaxf(c[v] + q2, 0.0f);  // restore |q|^2; clamp like max(d2,0)
      // Float prefilter: current 8th-smallest d2 is the hi word of kl[7]
      // (free register read). For non-negative floats, d2c < thr implies the
      // full u64 key beats kl[7], so the hit path needs no re-check.
      float thr = __uint_as_float((unsigned)(kl[KNN - 1] >> 32));
      if (d2c < thr) {
        u64 key = ((u64)__float_as_uint(d2c) << 32) | (unsigned)(mbase + v);
        insert8_hit(kl, key);
      }
    }

    av = nx;
  }

  // Merge partner half-list (lane ^ 16 covers the other 8 rows of each tile).
  // Snapshot via shfl first so both halves read pre-merge values; exact u64
  // guarded insert here (ties across halves resolved by lower index).
  unsigned ohi[KNN], olo[KNN];
#pragma unroll
  for (int j = 0; j < KNN; ++j) {
    ohi[j] = (unsigned)__shfl((int)(kl[j] >> 32), lane ^ 16, 32);
    olo[j] = (unsigned)__shfl((int)(kl[j] & 0xffffffffu), lane ^ 16, 32);
  }
#pragma unroll
  for (int j = 0; j < KNN; ++j) {
    insert8(kl, ((u64)ohi[j] << 32) | olo[j]);
  }

  // Lanes 0..15 now hold the global top-8 for their query.
  if (hi == 0) {
    float wsum = 0.f, fx = 0.f, fy = 0.f, fz = 0.f;
#pragma unroll
    for (int j = 0; j < KNN; ++j) {
      float d2 = __uint_as_float((unsigned)(kl[j] >> 32));
      int m = (int)(kl[j] & 0xffffffffu);
      float w = 1.0f / (d2 + 1e-8f);  // POWER==2 -> no sqrt needed
      wsum += w;
      fx += w * rflow[3 * m + 0];
      fy += w * rflow[3 * m + 1];
      fz += w * rflow[3 * m + 2];
    }
    float inv = 1.0f / wsum;
    out[3 * q + 0] = fx * inv;
    out[3 * q + 1] = fy * inv;
    out[3 * q + 2] = fz * inv;
  }
}

extern "C" void kernel_launch(void* const* d_in, const int* in_sizes, int n_in,
                              void* d_out, int out_size, void* d_ws, size_t ws_size,
                              hipStream_t stream) {
  const float* qpts  = (const float*)d_in[0];  // [N,3]
  const float* rpts  = (const float*)d_in[1];  // [M,3]
  const float* rflow = (const float*)d_in[2];  // [M,3]
  float* out = (float*)d_out;                  // [N,3]

  const int N = in_sizes[0] / 3;  // 32768
  const int M = in_sizes[1] / 3;  // 16384

  float4* pk = (float4*)d_ws;  // M * 16B = 256 KB scratch

  pack_refs_kernel<<<(M + 255) / 256, 256, 0, stream>>>(rpts, pk, M);

  const int waves  = N / 16;            // 2048 waves, 16 queries each
  const int blocks = (waves + 7) / 8;   // 256 threads = 8 waves per block
  // M/16 tiles + one 256B pad tile so the last prefetch stays in-bounds
  const size_t lds_bytes = (size_t)M * sizeof(float4) + 256;
  idw_knn_wmma_kernel<<<blocks, 256, lds_bytes, stream>>>(qpts, pk, rflow, out,
                                                          N, M);
}